// RotatedGridPool_70970039599822
// MI455X (gfx1250) — compile-verified
//
#include <hip/hip_runtime.h>
#include <cstdint>

// RotatedGridPool for MI455X (gfx1250, wave32).
// One 256-thread block per ROI (thread == channel). Per-point tap offsets and
// validity-folded bilinear weights are computed once into LDS by threads
// 0..48; each thread then gathers 4 taps per grid point from its channel
// plane (feature map is L2-resident: 144 MB < 192 MB L2) and writes the
// [C][49] output tile into LDS. The tile is streamed to global memory
// contiguously with gfx1250 async LDS->global b128 stores (ASYNCcnt path).

namespace {
constexpr int   kG    = 7;
constexpr int   kP    = kG * kG;   // 49 grid points
constexpr int   kC    = 256;
constexpr int   kH    = 200;
constexpr int   kW    = 176;
constexpr float kMinX = 0.0f;
constexpr float kMinY = -40.0f;
}  // namespace

__global__ __launch_bounds__(kC) void rotated_grid_pool_kernel(
    const float* __restrict__ feat,         // (B, C, H, W)
    const float* __restrict__ rois,         // (B, N, 7)
    const float* __restrict__ voxel,        // (2,)
    const int*   __restrict__ fmap_stride,  // (1,)
    float*       __restrict__ out,          // (B*N, C, G, G)
    int nroi_per_batch) {
  __shared__ __align__(16) float s_out[kC * kP];  // 50176 B, contiguous ROI tile
  __shared__ int   s_o0[kP];
  __shared__ int   s_o1[kP];
  __shared__ float s_wr0[kP];
  __shared__ float s_wr1[kP];
  __shared__ float s_wa[kP];
  __shared__ float s_wb[kP];

  const int t   = (int)threadIdx.x;                       // channel index
  const int b   = (int)blockIdx.y;                        // batch index
  const int roi = b * nroi_per_batch + (int)blockIdx.x;   // global ROI index

  // ---- Phase 1: per-point tap offsets & weights (threads 0..48) ----
  if (t < kP) {
    const float* rp = rois + (size_t)roi * 7;
    const float cx = rp[0], cy = rp[1], dx = rp[3], dy = rp[4], ang = rp[6];
    const float fs = (float)fmap_stride[0];
    const float vx = voxel[0] * fs;
    const float vy = voxel[1] * fs;
    const float x1 = (cx - 0.5f * dx - kMinX) / vx;
    const float x2 = (cx + 0.5f * dx - kMinX) / vx;
    const float y1 = (cy - 0.5f * dy - kMinY) / vy;
    const float y2 = (cy + 0.5f * dy - kMinY) / vy;
    const float ca = cosf(ang), sa = sinf(ang);
    const float ex = x2 - x1, ey = y2 - y1;
    const float scale1 = ey / fmaxf(ex, 0.01f);
    const float scale2 = ex / fmaxf(ey, 0.01f);
    const float iw = 1.0f / (float)(kW - 1);
    const float ih = 1.0f / (float)(kH - 1);
    const float t00 = ex * iw * ca;
    const float t01 = ex * iw * (-sa) * scale1;
    const float t02 = (x1 + x2 - (float)(kW - 1)) * iw;
    const float t10 = ey * ih * sa * scale2;
    const float t11 = ey * ih * ca;
    const float t12 = (y1 + y2 - (float)(kH - 1)) * ih;

    const int gxi = t % kG, gyi = t / kG;
    const float xx = (2.0f * (float)gxi + 1.0f) / (float)kG - 1.0f;
    const float yy = (2.0f * (float)gyi + 1.0f) / (float)kG - 1.0f;
    const float gx = t00 * xx + t01 * yy + t02;
    const float gy = t10 * xx + t11 * yy + t12;
    // grid_sample, bilinear, zeros padding, align_corners=False
    const float ix = ((gx + 1.0f) * (float)kW - 1.0f) * 0.5f;
    const float iy = ((gy + 1.0f) * (float)kH - 1.0f) * 0.5f;
    const float x0f = floorf(ix), y0f = floorf(iy);
    const float wx1 = ix - x0f, wx0 = 1.0f - wx1;
    const float wy1 = iy - y0f, wy0 = 1.0f - wy1;
    const int x0 = (int)x0f, y0 = (int)y0f;

    // Rows: clamp index, zero weight if invalid (matches reference masking).
    const int   r0  = y0 < 0 ? 0 : (y0 > kH - 1 ? kH - 1 : y0);
    const int   y1i = y0 + 1;
    const int   r1  = y1i < 0 ? 0 : (y1i > kH - 1 ? kH - 1 : y1i);
    const float wr0 = (y0 >= 0 && y0 <= kH - 1) ? wy0 : 0.0f;
    const float wr1 = (y1i >= 0 && y1i <= kH - 1) ? wy1 : 0.0f;
    // Columns: load the pair at xb,xb+1 with xb=clamp(x0,0,W-2); remap the
    // bilinear weights onto the loaded elements (weight 0 for non-taps).
    const int   xb = x0 < 0 ? 0 : (x0 > kW - 2 ? kW - 2 : x0);
    const float wa = (xb == x0) ? wx0 : ((xb == x0 + 1) ? wx1 : 0.0f);
    const float wb = (xb + 1 == x0) ? wx0 : ((xb == x0) ? wx1 : 0.0f);

    s_o0[t]  = r0 * kW + xb;
    s_o1[t]  = r1 * kW + xb;
    s_wr0[t] = wr0;
    s_wr1[t] = wr1;
    s_wa[t]  = wa;
    s_wb[t]  = wb;
  }
  __syncthreads();

  // ---- Phase 2: gather + bilinear combine, one channel per lane ----
  const float* plane = feat + (size_t)(b * kC + t) * (size_t)(kH * kW);
  float* srow = s_out + t * kP;
#pragma unroll 7
  for (int p = 0; p < kP; ++p) {
    const int   o0 = s_o0[p];
    const int   o1 = s_o1[p];
    const float wa = s_wa[p], wb = s_wb[p];
    const float v0 = wa * plane[o0] + wb * plane[o0 + 1];
    const float v1 = wa * plane[o1] + wb * plane[o1 + 1];
    srow[p] = s_wr0[p] * v0 + s_wr1[p] * v1;
  }
  __syncthreads();

  // ---- Phase 3: bulk contiguous store LDS -> global via CDNA5 async path ----
  // 50176 B per ROI = 3136 x b128; 512 B per wave-instruction, fully coalesced.
  const uint64_t gbase = (uint64_t)(uintptr_t)(out + (size_t)roi * (kC * kP));
  // Low 32 bits of an LDS flat address are the LDS byte offset (ISA §10.2).
  const uint32_t lbase = (uint32_t)(uintptr_t)(void*)&s_out[0];
  constexpr int kChunks = kC * kP * 4 / 16;  // 3136
  for (int idx = t; idx < kChunks; idx += kC) {
    const uint64_t ga = gbase + (uint64_t)idx * 16u;
    const uint32_t la = lbase + (uint32_t)idx * 16u;
    asm volatile("global_store_async_from_lds_b128 %0, %1, off"
                 :: "v"(ga), "v"(la)
                 : "memory");
  }
  asm volatile("s_wait_asynccnt 0x0" ::: "memory");
}

extern "C" void kernel_launch(void* const* d_in, const int* in_sizes, int n_in,
                              void* d_out, int out_size, void* d_ws, size_t ws_size,
                              hipStream_t stream) {
  (void)n_in; (void)out_size; (void)d_ws; (void)ws_size;
  const float* feat  = (const float*)d_in[0];
  const float* rois  = (const float*)d_in[1];
  const float* voxel = (const float*)d_in[2];
  const int*   fstrd = (const int*)d_in[3];
  float*       out   = (float*)d_out;

  const int B = in_sizes[0] / (kC * kH * kW);   // 4
  const int N = in_sizes[1] / (B * 7);          // 128

  rotated_grid_pool_kernel<<<dim3(N, B), dim3(kC), 0, stream>>>(
      feat, rois, voxel, fstrd, out, N);
}